// PairwiseRankingLoss_31464930410802
// MI455X (gfx1250) — compile-verified
//
#include <hip/hip_runtime.h>
#include <hip/hip_bf16.h>
#include <math.h>

#define EPS_F 1e-6f
#define V_DIM 128

typedef __attribute__((ext_vector_type(2))) float v2f;
typedef __attribute__((ext_vector_type(8))) float v8f;

// One block per batch row b. 128 threads = 4 wave32.
__global__ __launch_bounds__(V_DIM) void prl_main_kernel(
    const float* __restrict__ pred,
    const float* __restrict__ targ,
    const int*   __restrict__ mask,
    float* __restrict__ wsum,
    float* __restrict__ wcnt) {
  __shared__ float sp[V_DIM];
  __shared__ float st[V_DIM];
  __shared__ float parts[8];  // 4 waves x {loss, count}

  const int i = threadIdx.x;
  const int b = blockIdx.x;
  const int g = b * V_DIM + i;

  const float pv = pred[g];
  const float tv = targ[g];
  const int   mv = mask[g];

  sp[i] = pv;
  // Fold validity into t: NaN poisons dt so fabsf(dt) > EPS is false.
  st[i] = mv ? tv : __int_as_float(0x7fc00000);
  __syncthreads();

  const float ti = st[i];
  const float pi = pv;
  float s = 0.0f;
  float c = 0.0f;

  // Balanced pair schedule. The pair loss is symmetric under i<->j
  // (dt'=-dt, dp'=-dp => x'=x), so each unordered pair {i,(i+k)%128} is
  // processed once: k=1..63 by every thread, k=64 only by threads i<64
  // (wave-uniform cut: waves 0-1 do 64 trips, waves 2-3 do 63).
  const int kmax = (i < (V_DIM / 2)) ? (V_DIM / 2) : (V_DIM / 2 - 1);
  #pragma unroll 4
  for (int k = 1; k <= kmax; ++k) {
    const int   j  = (i + k) & (V_DIM - 1);
    const float dt = st[j] - ti;               // NaN if either visit invalid
    const bool  ok = fabsf(dt) > EPS_F;        // false for NaN
    const float dp = pi - sp[j];
    const float x  = (dt > 0.0f) ? -dp : dp;   // -dp * sign(dt)
    // stable softplus(x) = max(x,0) + log(1 + exp(-|x|))
    const float l  = fmaxf(x, 0.0f) + __logf(1.0f + __expf(-fabsf(x)));
    s += ok ? l : 0.0f;
    c += ok ? 1.0f : 0.0f;
  }

  // ---- Wave32 dual reduction via V_WMMA_F32_16X16X4_F32 ----
  // A (16x4): row m gets K0=s_m, K1=c_m (lanes 0-15) / K2=s_{m+16}, K3=c_{m+16}
  //           (lanes 16-31) per the ISA 32-bit A layout.
  // B (4x16): B[k][n] = 1 iff parity(k) == parity(n). Then
  //   D[m][even n] = s_m + s_{m+16},  D[m][odd n] = c_m + c_{m+16}.
  const int lane = i & 31;
  v2f a;  a.x = s;  a.y = c;
  v2f bb;
  bb.x = (lane & 1) ? 0.0f : 1.0f;   // rows K=0,2
  bb.y = (lane & 1) ? 1.0f : 0.0f;   // rows K=1,3
  v8f cm = {};
  v8f d = __builtin_amdgcn_wmma_f32_16x16x4_f32(
      /*neg_a=*/false, a, /*neg_b=*/false, bb,
      /*c_mod=*/(short)0, cm, /*reuse_a=*/false, /*reuse_b=*/false);

  // Lane L holds D[v][L%16] for M=v (lanes 0-15) or M=8+v (lanes 16-31).
  float tot = d[0] + d[1] + d[2] + d[3] + d[4] + d[5] + d[6] + d[7];
  tot += __shfl_xor(tot, 16, 32);
  // Now: even lanes = full-wave loss sum, odd lanes = full-wave count.

  const int wave = i >> 5;
  if (lane < 2) parts[wave * 2 + lane] = tot;
  __syncthreads();

  if (i == 0) {
    const float S = parts[0] + parts[2] + parts[4] + parts[6];
    const float C = parts[1] + parts[3] + parts[5] + parts[7];
    wsum[b] = S;
    wcnt[b] = C;
  }
}

// Deterministic fixed-order final reduction of per-block partials.
__global__ __launch_bounds__(256) void prl_finalize_kernel(
    const float* __restrict__ wsum,
    const float* __restrict__ wcnt,
    float* __restrict__ out,
    int n) {
  __shared__ float ls[256];
  __shared__ float lc[256];
  const int t = threadIdx.x;
  float s = 0.0f, c = 0.0f;
  for (int i = t; i < n; i += 256) {
    s += wsum[i];
    c += wcnt[i];
  }
  ls[t] = s;
  lc[t] = c;
  __syncthreads();
  for (int off = 128; off > 0; off >>= 1) {
    if (t < off) {
      ls[t] += ls[t + off];
      lc[t] += lc[t + off];
    }
    __syncthreads();
  }
  if (t == 0) {
    out[0] = (lc[0] > 0.0f) ? (ls[0] / lc[0]) : 0.0f;
  }
}

extern "C" void kernel_launch(void* const* d_in, const int* in_sizes, int n_in,
                              void* d_out, int out_size, void* d_ws, size_t ws_size,
                              hipStream_t stream) {
  (void)n_in; (void)out_size; (void)ws_size;
  const float* pred = (const float*)d_in[0];
  const float* targ = (const float*)d_in[1];
  const int*   mask = (const int*)d_in[2];

  const int B = in_sizes[0] / V_DIM;  // 2048 for the reference shapes

  float* wsum = (float*)d_ws;
  float* wcnt = wsum + B;

  prl_main_kernel<<<B, V_DIM, 0, stream>>>(pred, targ, mask, wsum, wcnt);
  prl_finalize_kernel<<<1, 256, 0, stream>>>(wsum, wcnt, (float*)d_out, B);
}